// DiffusionDecoder_24584392802793
// MI455X (gfx1250) — compile-verified
//
#include <hip/hip_runtime.h>
#include <hip/hip_bf16.h>
#include <math.h>

typedef __attribute__((ext_vector_type(2))) float v2f;
typedef __attribute__((ext_vector_type(8))) float v8f;

#define NB      2048          // crystals
#define NPER    64            // atoms per crystal
#define NTOT    (NB * NPER)   // 131072 nodes
#define NDIM    64            // node feature dim
#define HDIM    128           // hidden dim
#define NSPEC   100           // species classes
#define TSTEPS  1000

// ---------------- workspace layout (floats) ----------------
#define WS_SA    0            // 1000 sqrt(alpha_cumprod)
#define WS_SO    1024         // 1000 sqrt(1 - alpha_cumprod)
#define WS_PM    2048         // 512  mse partials
#define WS_PR    2560         // 2048 repulsion partials
#define WS_PS    4608         // 2048 species partials

// ---------------------------------------------------------------------------
// Kernel 0: build cosine diffusion schedule (matches numpy float64 reference)
// ---------------------------------------------------------------------------
__global__ void init_tables_kernel(float* __restrict__ sa, float* __restrict__ so) {
    if (threadIdx.x == 0 && blockIdx.x == 0) {
        const double s  = 0.008;
        const double pi = 3.14159265358979323846;
        double prev = cos((s / (1.0 + s)) * pi * 0.5);
        prev *= prev;                       // acp_raw[0]
        double ac = 1.0;                    // running cumprod of (1-beta)
        for (int i = 1; i <= TSTEPS; ++i) {
            double cur = cos((((double)i / (double)TSTEPS) + s) / (1.0 + s) * pi * 0.5);
            cur *= cur;
            double beta = 1.0 - cur / prev; // normalization by acp[0] cancels
            if (beta < 1e-4)  beta = 1e-4;
            if (beta > 0.999) beta = 0.999;
            ac *= (1.0 - beta);
            sa[i - 1] = (float)sqrt(ac);
            so[i - 1] = (float)sqrt(1.0 - ac);
            prev = cur;
        }
    }
}

// ---------------------------------------------------------------------------
// Kernel 1: sum of (pred_noise - noise)^2  -> per-block partials
// ---------------------------------------------------------------------------
__global__ __launch_bounds__(256) void mse_kernel(const float* __restrict__ pn,
                                                  const float* __restrict__ nz,
                                                  int n, float* __restrict__ part) {
    __shared__ float red[256];
    float s = 0.0f;
    for (int i = blockIdx.x * 256 + threadIdx.x; i < n; i += gridDim.x * 256) {
        float d = pn[i] - nz[i];
        s += d * d;
    }
    red[threadIdx.x] = s;
    __syncthreads();
    for (int off = 128; off > 0; off >>= 1) {
        if ((int)threadIdx.x < off) red[threadIdx.x] += red[threadIdx.x + off];
        __syncthreads();
    }
    if (threadIdx.x == 0) part[blockIdx.x] = red[0];
}

// ---------------------------------------------------------------------------
// Kernel 2: pred_x0 + periodic-boundary repulsion, one block per crystal
// ---------------------------------------------------------------------------
__global__ __launch_bounds__(64) void rep_kernel(const float* __restrict__ frac,
                                                 const float* __restrict__ nz,
                                                 const float* __restrict__ pn,
                                                 const float* __restrict__ lattice,
                                                 const int*   __restrict__ t,
                                                 const float* __restrict__ sa_t,
                                                 const float* __restrict__ so_t,
                                                 float* __restrict__ part) {
    __shared__ float f[NPER * 3];
    __shared__ float lat[9];
    __shared__ float red[NPER];

    const int b  = blockIdx.x;
    const int i  = threadIdx.x;
    const int tb = t[b];
    const float sa = sa_t[tb];
    const float so = so_t[tb];
    const int node = b * NPER + i;

    #pragma unroll
    for (int k = 0; k < 3; ++k) {
        float xt = sa * frac[node * 3 + k] + so * nz[node * 3 + k];
        xt -= floorf(xt);                               // wrap
        float p = (xt - so * pn[node * 3 + k]) / sa;    // predicted x0
        p -= floorf(p);                                 // wrap
        f[i * 3 + k] = p;
    }
    if (i < 9) lat[i] = lattice[b * 9 + i];
    __syncthreads();

    const float fx = f[i * 3 + 0], fy = f[i * 3 + 1], fz = f[i * 3 + 2];
    float acc = 0.0f;
    for (int j = 0; j < NPER; ++j) {
        float d0 = fx - f[j * 3 + 0];
        float d1 = fy - f[j * 3 + 1];
        float d2 = fz - f[j * 3 + 2];
        d0 -= rintf(d0); d1 -= rintf(d1); d2 -= rintf(d2);   // minimum image
        float c0 = d0 * lat[0] + d1 * lat[3] + d2 * lat[6];
        float c1 = d0 * lat[1] + d1 * lat[4] + d2 * lat[7];
        float c2 = d0 * lat[2] + d1 * lat[5] + d2 * lat[8];
        float dsq = c0 * c0 + c1 * c1 + c2 * c2;
        if (j == i) dsq = 1e10f;
        float dist = sqrtf(dsq + 1e-8f);
        float r = fmaxf(0.8f - dist, 0.0f);
        acc += r * r;
    }
    red[i] = acc;
    __syncthreads();
    for (int off = 32; off > 0; off >>= 1) {
        if (i < off) red[i] += red[i + off];
        __syncthreads();
    }
    if (i == 0) part[b] = red[0];
}

// ---------------------------------------------------------------------------
// Kernel 3: species head via fp32 WMMA (V_WMMA_F32_16X16X4_F32)
//   Linear(64->128) + SiLU + Linear(128->100) + log_softmax + NLL
//   128 threads = 4 waves; each wave owns a 16-row tile.
// ---------------------------------------------------------------------------
#define HID_STRIDE 132        // 128 + 4 pad -> conflict-free LDS
#define LG_STRIDE  116        // 112 + 4 pad (cols padded 100 -> 112)

__global__ __launch_bounds__(128) void species_kernel(const float* __restrict__ h_final,
                                                      const float* __restrict__ W1,
                                                      const float* __restrict__ b1,
                                                      const float* __restrict__ W2,
                                                      const float* __restrict__ b2,
                                                      const int*   __restrict__ species,
                                                      float* __restrict__ part) {
    __shared__ float hid_s[4 * 16 * HID_STRIDE];   // per-wave 16x128 hidden
    __shared__ float lg_s [4 * 16 * LG_STRIDE];    // per-wave 16x112 logits
    __shared__ float rowloss[64];

    const int lane = threadIdx.x & 31;
    const int wid  = threadIdx.x >> 5;
    const int row  = lane & 15;       // M index (also N index for B/C layout)
    const int kh   = lane >> 4;       // K-half selector for A/B layout
    const int r0   = blockIdx.x * 64 + wid * 16;   // first global row of tile

    float* hid = hid_s + wid * 16 * HID_STRIDE;
    float* lg  = lg_s  + wid * 16 * LG_STRIDE;

    // ---- preload A tile (16x64 of h_final) in WMMA A-layout registers ----
    // lane<16 : a[ks] = A[row][4ks+0], A[row][4ks+1]
    // lane>=16: a[ks] = A[row][4ks+2], A[row][4ks+3]
    v2f a[16];
    const float* Arow = h_final + (size_t)(r0 + row) * NDIM;
    #pragma unroll
    for (int ks = 0; ks < 16; ++ks) {
        int k = ks * 4 + kh * 2;
        a[ks].x = Arow[k];
        a[ks].y = Arow[k + 1];
    }

    // ---- GEMM1: hidden = silu(h @ W1 + b1), N-tile at a time ----
    for (int nt = 0; nt < HDIM / 16; ++nt) {
        const int col = nt * 16 + row;
        const float bb = b1[col];
        v8f c;
        #pragma unroll
        for (int v = 0; v < 8; ++v) c[v] = bb;
        #pragma unroll
        for (int ks = 0; ks < 16; ++ks) {
            int k = ks * 4 + kh * 2;
            v2f bvec;
            bvec.x = W1[k * HDIM + col];
            bvec.y = W1[(k + 1) * HDIM + col];
            c = __builtin_amdgcn_wmma_f32_16x16x4_f32(false, a[ks], false, bvec,
                                                      (short)0, c, false, false);
        }
        // C layout: VGPR v holds M=v (lanes 0-15) / M=v+8 (lanes 16-31), N=col
        #pragma unroll
        for (int v = 0; v < 8; ++v) {
            float x = c[v];
            float sv = x / (1.0f + expf(-x));          // SiLU
            hid[(v + 8 * kh) * HID_STRIDE + col] = sv;
        }
    }
    __syncthreads();

    // ---- GEMM2: logits = hidden @ W2 + b2 (cols padded to 112) ----
    for (int nt = 0; nt < 7; ++nt) {
        const int col = nt * 16 + row;
        const bool ok = (col < NSPEC);
        const float bb = ok ? b2[col] : 0.0f;
        v8f c;
        #pragma unroll
        for (int v = 0; v < 8; ++v) c[v] = bb;
        #pragma unroll
        for (int ks = 0; ks < 32; ++ks) {
            int k = ks * 4 + kh * 2;
            v2f aa, bvec;
            aa.x = hid[row * HID_STRIDE + k];
            aa.y = hid[row * HID_STRIDE + k + 1];
            bvec.x = ok ? W2[k * NSPEC + col]       : 0.0f;
            bvec.y = ok ? W2[(k + 1) * NSPEC + col] : 0.0f;
            c = __builtin_amdgcn_wmma_f32_16x16x4_f32(false, aa, false, bvec,
                                                      (short)0, c, false, false);
        }
        #pragma unroll
        for (int v = 0; v < 8; ++v)
            lg[(v + 8 * kh) * LG_STRIDE + col] = c[v];
    }
    __syncthreads();

    // ---- log_softmax + NLL over the 100 real columns ----
    if (lane < 16) {
        const float* lr = lg + lane * LG_STRIDE;
        float m = -3.4e38f;
        for (int c2 = 0; c2 < NSPEC; ++c2) m = fmaxf(m, lr[c2]);
        float ssum = 0.0f;
        for (int c2 = 0; c2 < NSPEC; ++c2) ssum += expf(lr[c2] - m);
        const int sp = species[r0 + lane];
        const float lp = lr[sp] - m - logf(ssum);
        rowloss[wid * 16 + lane] = -lp;
    }
    __syncthreads();
    if (threadIdx.x == 0) {
        float s = 0.0f;
        #pragma unroll
        for (int i = 0; i < 64; ++i) s += rowloss[i];
        part[blockIdx.x] = s;
    }
}

// ---------------------------------------------------------------------------
// Kernel 4: deterministic final reduction -> 3 scalar outputs
// ---------------------------------------------------------------------------
__global__ __launch_bounds__(256) void finalize_kernel(const float* __restrict__ pm, int nm,
                                                       const float* __restrict__ pr, int nr,
                                                       const float* __restrict__ ps, int ns,
                                                       float* __restrict__ out) {
    __shared__ float r0[256], r1[256], r2[256];
    const int tid = threadIdx.x;
    float s0 = 0.0f, s1 = 0.0f, s2 = 0.0f;
    for (int i = tid; i < nm; i += 256) s0 += pm[i];
    for (int i = tid; i < nr; i += 256) s1 += pr[i];
    for (int i = tid; i < ns; i += 256) s2 += ps[i];
    r0[tid] = s0; r1[tid] = s1; r2[tid] = s2;
    __syncthreads();
    for (int off = 128; off > 0; off >>= 1) {
        if (tid < off) { r0[tid] += r0[tid + off]; r1[tid] += r1[tid + off]; r2[tid] += r2[tid + off]; }
        __syncthreads();
    }
    if (tid == 0) {
        float mse  = r0[0] / (3.0f * (float)NTOT);
        float lrep = r1[0] / ((float)NPER * (float)NB);
        out[0] = mse + 5.0f * lrep;          // loss_diffusion
        out[1] = r2[0] / (float)NTOT;        // loss_species
        out[2] = lrep;                       // l_repulsion
    }
}

// ---------------------------------------------------------------------------
extern "C" void kernel_launch(void* const* d_in, const int* in_sizes, int n_in,
                              void* d_out, int out_size, void* d_ws, size_t ws_size,
                              hipStream_t stream) {
    const float* frac    = (const float*)d_in[0];
    const float* noise   = (const float*)d_in[1];
    const float* pnoise  = (const float*)d_in[2];
    const float* h_final = (const float*)d_in[3];
    const float* lattice = (const float*)d_in[4];
    const float* W1      = (const float*)d_in[5];
    const float* b1      = (const float*)d_in[6];
    const float* W2      = (const float*)d_in[7];
    const float* b2      = (const float*)d_in[8];
    const int*   t       = (const int*)d_in[9];
    // d_in[10] = batch_indices (uniform groups -> derived analytically)
    const int*   species = (const int*)d_in[11];
    float* out = (float*)d_out;

    float* ws = (float*)d_ws;
    float* sa = ws + WS_SA;
    float* so = ws + WS_SO;
    float* pm = ws + WS_PM;
    float* pr = ws + WS_PR;
    float* ps = ws + WS_PS;

    init_tables_kernel<<<1, 1, 0, stream>>>(sa, so);
    mse_kernel<<<512, 256, 0, stream>>>(pnoise, noise, NTOT * 3, pm);
    rep_kernel<<<NB, NPER, 0, stream>>>(frac, noise, pnoise, lattice, t, sa, so, pr);
    species_kernel<<<NTOT / 64, 128, 0, stream>>>(h_final, W1, b1, W2, b2, species, ps);
    finalize_kernel<<<1, 256, 0, stream>>>(pm, 512, pr, NB, ps, NTOT / 64, out);
}